// MemEfficient_FerroelectricBasisConv2d_87978110091303
// MI455X (gfx1250) — compile-verified
//
#include <hip/hip_runtime.h>
#include <cstdint>
#include <cstddef>

// ---------------------------------------------------------------------------
// FerroelectricBasisConv2d forward, MI455X (gfx1250).
// Forward-pass algebra: is_up==0.5 exactly, crossed_pos cancels, so
//   out[b,co,p] = sum_j coef*Ps*tanh(k*((x+Ec) - 0.2*Ec*sigmoid(-10(x+Ec))))
//               + sum_j coef*bias + out_bias[co]
// 56.6M inner elements x {exp2, rcp, tanh} -> transcendental-bound; total
// memory ~1MB (L2-resident, 192MB L2; 23.3TB/s HBM irrelevant). No bilinear
// A*B structure (params enter through the nonlinearity) -> WMMA inapplicable.
// CDNA5 features: TDM tensor_load_to_lds + s_wait_tensorcnt for LDS staging,
// v_exp_f32 / v_rcp_f32 / v_tanh_f32, wave32 blocks, LDS broadcast reads.
// ---------------------------------------------------------------------------

#define B_      4
#define CIN     16
#define COUT    32
#define HH      32
#define WW      32
#define NB_     3
#define NPX     (HH * WW)          // 1024
#define NPARAM  (CIN * NB_ * 9)    // 432 per cout
#define LOG2E   1.4426950408889634f

// ---- fast-math dispatch (amdgcn builtins only exist in the device pass) ----
#if defined(__HIP_DEVICE_COMPILE__)
  #define FAST_EXP2(x) __builtin_amdgcn_exp2f(x)
  #define FAST_RCP(x)  __builtin_amdgcn_rcpf(x)
  #if __has_builtin(__builtin_amdgcn_tanhf)
    #define FAST_TANH(x) __builtin_amdgcn_tanhf(x)   // gfx1250 v_tanh_f32
  #endif
#else
  #define FAST_EXP2(x) exp2f(x)
  #define FAST_RCP(x)  (1.0f / (x))
#endif

#ifndef FAST_TANH
// tanh(u) = 1 - 2/(exp(2u)+1), via native exp2 + rcp
__device__ __host__ inline float fast_tanh_fallback(float u) {
  float eu = FAST_EXP2((2.0f * LOG2E) * u);
  return fmaf(-2.0f, FAST_RCP(eu + 1.0f), 1.0f);
}
  #define FAST_TANH(x) fast_tanh_fallback(x)
#endif

// ---- TDM availability ------------------------------------------------------
#if defined(__HIP_DEVICE_COMPILE__) && defined(__gfx1250__)
  #if __has_builtin(__builtin_amdgcn_tensor_load_to_lds) && \
      __has_builtin(__builtin_amdgcn_s_wait_tensorcnt)
    #define HAVE_TDM 1
  #endif
#endif
#if __has_include(<hip/amd_detail/amd_gfx1250_TDM.h>)
  #define TDM_SIX_ARG 1   // amdgpu-toolchain clang-23: 6-arg builtin
#endif

typedef __attribute__((ext_vector_type(4))) unsigned int u32x4;
typedef __attribute__((ext_vector_type(4))) int          i32x4;
typedef __attribute__((ext_vector_type(8))) int          i32x8;

#ifdef HAVE_TDM
// 1-D contiguous TDM load: n4 dwords from global -> LDS byte offset lds_off.
// D# per CDNA5 ISA: group0 = {count|gather, lds_addr,
// global_addr[56:0], type=2}; group1 packs data_size/dims/strides.
__device__ inline void tdm_load_1d(uint32_t lds_off, const void* gptr, uint32_t n4) {
  uint64_t ga = (uint64_t)(uintptr_t)gptr;
  u32x4 g0;
  g0.x = 1u;                                                  // count=1, user mode
  g0.y = lds_off;                                             // lds_addr (bytes)
  g0.z = (uint32_t)ga;                                        // global_addr[31:0]
  g0.w = ((uint32_t)(ga >> 32) & 0x01FFFFFFu) | 0x80000000u;  // addr[56:32] | type=2
  i32x8 g1;
  g1[0] = (int)0x00020000u;               // wg_mask=0, data_size=2 (4B), no pad/iter
  g1[1] = (int)(n4 << 16);                // tensor_dim0[15:0] @ bits 63:48
  g1[2] = (int)((n4 >> 16) | (1u << 16)); // tensor_dim0[31:16] | tensor_dim1=1 lo
  g1[3] = (int)(n4 << 16);                // tensor_dim1 hi=0 | tile_dim0=n4 (<=65535)
  g1[4] = 1;                              // tile_dim1=1, tile_dim2=0
  g1[5] = (int)n4;                        // tensor_dim0_stride lo32
  g1[6] = 0;                              // stride0 hi | stride1 lo
  g1[7] = 0;
  i32x4 gz4 = (i32x4)0;                   // groups 2/3: dims beyond 2D unused
#ifdef TDM_SIX_ARG
  i32x8 gz8 = (i32x8)0;
  __builtin_amdgcn_tensor_load_to_lds(g0, g1, gz4, gz4, gz8, 0);
#else
  __builtin_amdgcn_tensor_load_to_lds(g0, g1, gz4, gz4, 0);
#endif
}
#endif

// ---------------------------------------------------------------------------
// Main (defined FIRST so it heads the disasm snippet).
// Block = (b, co, pixel segment of 256). LDS: full x[b] (64KB) + this co's
// packed params (6.9KB) = 72.4KB/WG (~4 WGs per 320KB WGP, 8 wave32s each).
// Packed param float4 = {k, Ec, coef*Ps, -0.2*Ec}.
// ---------------------------------------------------------------------------
__global__ __launch_bounds__(256)
void ferro_fwd(const float* __restrict__ x, const float4* __restrict__ prm,
               const float* __restrict__ cconst, float* __restrict__ out) {
  __shared__ float  lds_x[CIN * NPX];   // 65536 B
  __shared__ float4 lds_p[NPARAM];      //  6912 B

  const int blk = blockIdx.x;
  const int seg = blk & 3;
  const int co  = (blk >> 2) & (COUT - 1);
  const int b   = blk >> 7;
  const int tid = threadIdx.x;

  const float*  xg = x   + (size_t)b  * (CIN * NPX);
  const float4* pg = prm + (size_t)co * NPARAM;

#ifdef HAVE_TDM
  if (tid < 32) {  // one wave issues the DMA pair; TDM ops are in-order per wave
    tdm_load_1d((uint32_t)(uintptr_t)&lds_x[0], xg, CIN * NPX);   // 16384 dw
    tdm_load_1d((uint32_t)(uintptr_t)&lds_p[0], pg, NPARAM * 4);  //  1728 dw
    __builtin_amdgcn_s_wait_tensorcnt(0);
  }
#else
  for (int i = tid; i < CIN * NPX; i += 256) lds_x[i] = xg[i];
  for (int i = tid; i < NPARAM;    i += 256) lds_p[i] = pg[i];
#endif
  __syncthreads();

  const int p = seg * 256 + tid;     // pixel id
  const int h = p >> 5;
  const int w = p & 31;

  float acc = 0.0f;
  #pragma unroll 1
  for (int ci = 0; ci < CIN; ++ci) {
    const float* xc = &lds_x[ci * NPX];
    float xv[9];                     // 3x3 patch with zero padding
    #pragma unroll
    for (int dy = 0; dy < 3; ++dy) {
      #pragma unroll
      for (int dx = 0; dx < 3; ++dx) {
        const int hh2 = h + dy - 1;
        const int ww2 = w + dx - 1;
        const bool ok = (hh2 >= 0) && (hh2 < HH) && (ww2 >= 0) && (ww2 < WW);
        xv[dy * 3 + dx] = ok ? xc[hh2 * WW + ww2] : 0.0f;
      }
    }
    const float4* pc = &lds_p[ci * (NB_ * 9)];   // uniform addr -> LDS broadcast
    #pragma unroll
    for (int nb = 0; nb < NB_; ++nb) {
      #pragma unroll
      for (int t = 0; t < 9; ++t) {
        const float4 q = pc[nb * 9 + t];          // {k, Ec, coef*Ps, -0.2*Ec}
        const float t1 = xv[t] + q.y;             // x + Ec
        const float e  = FAST_EXP2((10.0f * LOG2E) * t1);
        const float s  = FAST_RCP(1.0f + e);      // sigmoid(-10*(x+Ec))
        const float sh = fmaf(q.w, s, t1);        // (x+Ec) - 0.2*Ec*s
        const float th = FAST_TANH(q.x * sh);
        acc = fmaf(q.z, th, acc);
      }
    }
  }
  out[((size_t)b * COUT + co) * NPX + p] = acc + cconst[co];
}

// ---------------------------------------------------------------------------
// Prep: pack {k, Ec, coef*Ps, -0.2*Ec} and reduce coef*bias (+out_bias) per co.
// ws layout: float4 prm[COUT*NPARAM], then float cconst[COUT].
// ---------------------------------------------------------------------------
__global__ __launch_bounds__(256)
void ferro_prep(const float* __restrict__ k,  const float* __restrict__ Ec,
                const float* __restrict__ Ps, const float* __restrict__ bias,
                const float* __restrict__ coef, const float* __restrict__ out_bias,
                float4* __restrict__ prm, float* __restrict__ cconst) {
  const int co  = blockIdx.x;
  const int tid = threadIdx.x;
  __shared__ float red[256];
  float part = 0.0f;
  for (int i = tid; i < NPARAM; i += 256) {
    const int g = co * NPARAM + i;
    const float c = coef[g];
    const float e = Ec[g];
    prm[g] = make_float4(k[g], e, c * Ps[g], -0.2f * e);
    part += c * bias[g];
  }
  red[tid] = part;
  __syncthreads();
  for (int s = 128; s > 0; s >>= 1) {
    if (tid < s) red[tid] += red[tid + s];
    __syncthreads();
  }
  if (tid == 0) cconst[co] = red[0] + out_bias[co];
}

// ---------------------------------------------------------------------------
extern "C" void kernel_launch(void* const* d_in, const int* in_sizes, int n_in,
                              void* d_out, int out_size, void* d_ws, size_t ws_size,
                              hipStream_t stream) {
  const float* x        = (const float*)d_in[0];
  const float* k        = (const float*)d_in[1];
  const float* Ec       = (const float*)d_in[2];
  const float* Ps       = (const float*)d_in[3];
  const float* bias     = (const float*)d_in[4];
  const float* coef     = (const float*)d_in[5];
  const float* out_bias = (const float*)d_in[6];
  float* out = (float*)d_out;

  float4* prm   = (float4*)d_ws;                                   // 221184 B
  float*  ccons = (float*)((char*)d_ws + (size_t)COUT * NPARAM * sizeof(float4));

  ferro_prep<<<COUT, 256, 0, stream>>>(k, Ec, Ps, bias, coef, out_bias, prm, ccons);
  ferro_fwd<<<B_ * COUT * 4, 256, 0, stream>>>(x, prm, ccons, out);
}